// DoRALayer_65369402245693
// MI455X (gfx1250) — compile-verified
//
#include <hip/hip_runtime.h>

// DoRA forward, re-associated to minimize FLOPs and HBM traffic:
//   out = (||mag|| * alpha/D) * x @ V^T,  V = ((dir/||dir||_row) @ A^T) @ B^T
// => two D^3 GEMMs (8.6 GMAC each) + one N*D^2 GEMM (34.4 GMAC), ~103 GFLOP.
// Memory-bound (~180MB @ 23.3TB/s ~ 8us >> ~1us fp32-WMMA compute), so we use
// the native fp32 matrix path V_WMMA_F32_16X16X4_F32 (no precision loss) and
// spend the effort on data movement: GLOBAL_LOAD_ASYNC_TO_LDS_B128 (ASYNCcnt)
// with double-buffered LDS so the next tile streams in while WMMAs run.

typedef __attribute__((ext_vector_type(2))) float v2f;
typedef __attribute__((ext_vector_type(8))) float v8f;
typedef __attribute__((ext_vector_type(4))) int   v4i;

#define D_DIM   2048
#define N_TOK   8192
#define DORA_EPS 1e-12f
#define DORA_SCALING (16.0f / 2048.0f)

#define AS3 __attribute__((address_space(3)))

#if __has_builtin(__builtin_amdgcn_global_load_async_to_lds_b128)
#define USE_ASYNC_COPY 1
#else
#define USE_ASYNC_COPY 0
#endif

__device__ __forceinline__ void dora_wait_async_le8() {
#if __has_builtin(__builtin_amdgcn_s_wait_asynccnt)
  __builtin_amdgcn_s_wait_asynccnt(8);
#else
  asm volatile("s_wait_asynccnt 0x8" ::: "memory");
#endif
}
__device__ __forceinline__ void dora_wait_async_0() {
#if __has_builtin(__builtin_amdgcn_s_wait_asynccnt)
  __builtin_amdgcn_s_wait_asynccnt(0);
#else
  asm volatile("s_wait_asynccnt 0x0" ::: "memory");
#endif
}

// ---------------- reductions ----------------

__global__ void dora_rownorm_kernel(const float* __restrict__ dir,
                                    float* __restrict__ invn, int K) {
  __shared__ float red[256];
  const int row = blockIdx.x;
  const float* p = dir + (size_t)row * K;
  float s = 0.f;
  for (int i = threadIdx.x; i < K; i += 256) { float v = p[i]; s += v * v; }
  red[threadIdx.x] = s;
  __syncthreads();
  for (int off = 128; off > 0; off >>= 1) {
    if (threadIdx.x < off) red[threadIdx.x] += red[threadIdx.x + off];
    __syncthreads();
  }
  if (threadIdx.x == 0) invn[row] = 1.0f / fmaxf(sqrtf(red[0]), DORA_EPS);
}

__global__ void dora_magscale_kernel(const float* __restrict__ mag,
                                     float* __restrict__ out, int K) {
  __shared__ float red[256];
  float s = 0.f;
  for (int i = threadIdx.x; i < K; i += 256) { float v = mag[i]; s += v * v; }
  red[threadIdx.x] = s;
  __syncthreads();
  for (int off = 128; off > 0; off >>= 1) {
    if (threadIdx.x < off) red[threadIdx.x] += red[threadIdx.x + off];
    __syncthreads();
  }
  if (threadIdx.x == 0) out[0] = sqrtf(red[0]) * DORA_SCALING;
}

// ---------------- NT GEMM on fp32 WMMA ----------------
// C[M,N] = gscale * diag(rowScaleA) * Arow[M,K] * Brow[N,K]^T
// (rowScaleA applied in the epilogue -> staging is a pure copy, async-friendly)
// M,N,K multiples of 128. Block tile 128x128x32, 256 threads = 8 wave32s in a
// 4(M)x2(N) wave grid; each wave: 32x64 sub-tile = 2x4 WMMA accs (64 VGPRs).
// LDS stride 36 dwords: gcd(36,64)=4, 9 coprime 16 -> conflict-free b64 reads,
// and 36 dwords = 144B keeps every 4-dword chunk 16B-aligned for b128 copies.

#define BM 128
#define BN 128
#define BK 32
#define LDT (BK + 4)

__device__ __forceinline__ void dora_stage_tile(const float* __restrict__ G,
                                                float* L, int K,
                                                int grow0, int k0,
                                                int lr, int lc) {
  // 256 threads x 4 passes x 16B = 16KB = one 128x32 fp32 tile
  #pragma unroll
  for (int p = 0; p < 4; ++p) {
    const int row = lr + p * 32;
    const float* g = G + (size_t)(grow0 + row) * K + k0 + lc;
    float* l = L + row * LDT + lc;
#if USE_ASYNC_COPY
    __builtin_amdgcn_global_load_async_to_lds_b128(
        (v4i*)g, (AS3 v4i*)l, /*offset=*/0, /*cpol=*/0);
#else
    *reinterpret_cast<float4*>(l) = *reinterpret_cast<const float4*>(g);
#endif
  }
}

__global__ void __launch_bounds__(256)
dora_gemm_nt_wmma(const float* __restrict__ A, const float* __restrict__ B,
                  float* __restrict__ C, int M, int N, int K,
                  const float* __restrict__ rowScaleA,
                  const float* __restrict__ globalScale) {
  __shared__ float As[2][BM * LDT];
  __shared__ float Bs[2][BN * LDT];

  const int m0 = blockIdx.x * BM;
  const int n0 = blockIdx.y * BN;

  const int tid  = threadIdx.x;
  const int lane = tid & 31;
  const int wave = tid >> 5;
  const int wm   = wave & 3;          // 4 waves along M (32 rows each)
  const int wn   = wave >> 2;         // 2 waves along N (64 cols each)
  const int lm   = lane & 15;
  const int kh   = (lane >> 4) * 2;   // lanes 0-15 -> K0..1, lanes 16-31 -> K2..3

  v8f acc[2][4] = {};

  const int lr = tid >> 3;            // 0..31 base row for staging
  const int lc = (tid & 7) * 4;       // float4 column inside BK

  // prologue: start streaming tile 0 into buffer 0
  dora_stage_tile(A, As[0], K, m0, 0, lr, lc);
  dora_stage_tile(B, Bs[0], K, n0, 0, lr, lc);

  int cur = 0;
  for (int k0 = 0; k0 < K; k0 += BK) {
    const bool has_next = (k0 + BK) < K;
    if (has_next) {
      // stream tile k0+BK into the other buffer while we compute this one
      dora_stage_tile(A, As[cur ^ 1], K, m0, k0 + BK, lr, lc);
      dora_stage_tile(B, Bs[cur ^ 1], K, n0, k0 + BK, lr, lc);
      dora_wait_async_le8();   // my 8 ops for tile k0 done (in-order completion)
    } else {
      dora_wait_async_0();
    }
    __syncthreads();           // everyone's tile-k0 data visible in LDS

    #pragma unroll
    for (int k4 = 0; k4 < BK; k4 += 4) {
      v2f afrag[2], bfrag[4];
      #pragma unroll
      for (int i = 0; i < 2; ++i)
        afrag[i] = *reinterpret_cast<const v2f*>(
            &As[cur][(wm * 32 + i * 16 + lm) * LDT + k4 + kh]);
      #pragma unroll
      for (int j = 0; j < 4; ++j)
        bfrag[j] = *reinterpret_cast<const v2f*>(
            &Bs[cur][(wn * 64 + j * 16 + lm) * LDT + k4 + kh]);

      #pragma unroll
      for (int i = 0; i < 2; ++i)
        #pragma unroll
        for (int j = 0; j < 4; ++j)
          acc[i][j] = __builtin_amdgcn_wmma_f32_16x16x4_f32(
              false, afrag[i], false, bfrag[j],
              (short)0, acc[i][j], false, false);
    }
    __syncthreads();           // reads of buf[cur] done before it is re-filled
    cur ^= 1;
  }

  // epilogue: C/D layout -> VGPR g holds M = g + 8*(lane/16), N = lane%16
  const float gscale = globalScale ? globalScale[0] : 1.0f;
  const int hi16 = lane >> 4;
  #pragma unroll
  for (int i = 0; i < 2; ++i) {
    const int rowb = m0 + wm * 32 + i * 16 + hi16 * 8;
    float rs[8];
    #pragma unroll
    for (int g = 0; g < 8; ++g)
      rs[g] = (rowScaleA ? rowScaleA[rowb + g] : 1.0f) * gscale;
    #pragma unroll
    for (int j = 0; j < 4; ++j) {
      const int col = n0 + wn * 64 + j * 16 + lm;
      #pragma unroll
      for (int g = 0; g < 8; ++g)
        C[(size_t)(rowb + g) * N + col] = acc[i][j][g] * rs[g];
    }
  }
}

// ---------------- launch ----------------

extern "C" void kernel_launch(void* const* d_in, const int* in_sizes, int n_in,
                              void* d_out, int out_size, void* d_ws, size_t ws_size,
                              hipStream_t stream) {
  const float* x   = (const float*)d_in[0];  // [N_TOK, D]
  const float* lA  = (const float*)d_in[1];  // [D(rank), D(in)]
  const float* lB  = (const float*)d_in[2];  // [D(out), D(rank)]
  const float* mag = (const float*)d_in[3];  // [D]
  const float* dir = (const float*)d_in[4];  // [D(out), D(rank)]
  float* out = (float*)d_out;                // [N_TOK, D]

  const int D = D_DIM, Ntok = N_TOK;

  // workspace: [invn: D floats][msc: 1 float]; P (D*D) at byte 4096 overlaps invn/msc; V (D*D) after P ~32MB
  float* invn = (float*)d_ws;
  float* msc  = invn + D;
  float* P    = (float*)((char*)d_ws + 4096);
  float* V    = P + (size_t)D * D;

  dora_rownorm_kernel<<<D, 256, 0, stream>>>(dir, invn, D);
  dora_magscale_kernel<<<1, 256, 0, stream>>>(mag, msc, D);

  dim3 blk(256);
  dim3 g1(D / BM, D / BN);
  // P[o,r] = invn[o] * sum_i dir[o,i] * lA[r,i]   (invn applied in epilogue)
  dora_gemm_nt_wmma<<<g1, blk, 0, stream>>>(dir, lA, P, D, D, D, invn, nullptr);
  // V[o,p] = sum_r P[o,r] * lB[p,r]
  dora_gemm_nt_wmma<<<g1, blk, 0, stream>>>(P, lB, V, D, D, D, nullptr, nullptr);
  // out[n,o] = magscale * sum_p x[n,p] * V[o,p]
  dim3 g3(Ntok / BM, D / BN);
  dora_gemm_nt_wmma<<<g3, blk, 0, stream>>>(x, V, out, Ntok, D, D, nullptr, msc);
}